// AxialAttention_12128987644285
// MI455X (gfx1250) — compile-verified
//
#include <hip/hip_runtime.h>
#include <hip/hip_bf16.h>
#include <cstddef>

// ---------------------------------------------------------------------------
// AxialAttention for MI455X (gfx1250).  All matmuls use V_WMMA_F32_16X16X4_F32
// (native fp32 matrix pipe, wave32).  RPE terms are restructured into GEMMs
// (QR = Q*emb_q^T gather trick; wbar scatter + wbar*emb_v GEMM) so the matrix
// units do the heavy lifting instead of VALU gathers.  LDS tile staging uses
// the CDNA5 async-to-LDS path (ASYNCcnt) when the toolchain exposes it.
// Workspace requirement: ~235 MB (see layout below).
// ---------------------------------------------------------------------------

#define BB 4
#define LL 1024
#define CC 512
#define DKX 512
#define DVX 1024
#define HH 8
#define DKH 64
#define DVH 128
#define NQKV 2048          // 2*DK + DV
#define NREL 509           // 2*255 - 1
#define NRELP 512          // padded to multiple of 16
#define BL  (BB*LL)        // 4096 rows
#define BHL (BB*HH*LL)     // 32768 head-rows
#define EPSV 1e-5f

typedef __attribute__((ext_vector_type(2))) float v2f;
typedef __attribute__((ext_vector_type(8))) float v8f;
typedef int v4i __attribute__((vector_size(16)));

__device__ __forceinline__ v8f wmma4(v2f a, v2f b, v8f c) {
  return __builtin_amdgcn_wmma_f32_16x16x4_f32(false, a, false, b, (short)0, c,
                                               false, false);
}

// ---- CDNA5 async global->LDS staging (falls back to sync copy if absent) ----
#if defined(__has_builtin)
#if __has_builtin(__builtin_amdgcn_global_load_async_to_lds_b128) && \
    __has_builtin(__builtin_amdgcn_s_wait_asynccnt)
#define HAVE_ASYNC_LDS 1
#endif
#endif

typedef const __attribute__((address_space(1))) char* gas_cp;
typedef __attribute__((address_space(3))) char* las_p;
typedef __attribute__((address_space(1))) v4i* gas_v4i;
typedef __attribute__((address_space(3))) v4i* las_v4i;

// Copy `count` floats (count % 4 == 0, 16B aligned) from global to LDS.
// Issues global_load_async_to_lds_b128 per lane; waits own ASYNCcnt.
// Caller must __syncthreads() afterwards for cross-wave visibility.
__device__ __forceinline__ void stage_to_lds(const float* __restrict__ g,
                                             float* __restrict__ l, int count,
                                             int tid, int nthreads) {
#ifdef HAVE_ASYNC_LDS
  gas_cp gp = (gas_cp)g;
  las_p lp = (las_p)l;
  const int vecs = count >> 2;
  for (int i = tid; i < vecs; i += nthreads) {
    __builtin_amdgcn_global_load_async_to_lds_b128(
        (gas_v4i)(gp + 16 * i), (las_v4i)(lp + 16 * i), 0, 0);
  }
  __builtin_amdgcn_s_wait_asynccnt(0);
#else
  for (int i = tid; i < count; i += nthreads) l[i] = g[i];
#endif
}

// -------------------------------------------------------------- zero accums
__global__ void aa_zero_acc(float* sim_acc, float* out_acc) {
  int t = blockIdx.x * 256 + threadIdx.x;
  if (t < 48) sim_acc[t] = 0.f;
  if (t < 4096) out_acc[t] = 0.f;
}

// ----------------------------------------------- QKV GEMM: [4096,512]x[512,2048]
// One wave per 16x16 output tile.  grid = (2048/16, 4096/16), block = 32.
__global__ void aa_gemm_qkv(const float* __restrict__ x,
                            const float* __restrict__ W,
                            float* __restrict__ qkv) {
  const int lane = threadIdx.x & 31;
  const int hi = lane >> 4, lo = lane & 15;
  const int n0 = blockIdx.x * 16;
  const int m0 = blockIdx.y * 16;
  const float* arow = x + (size_t)(m0 + lo) * CC;
  v8f c = {0.f, 0.f, 0.f, 0.f, 0.f, 0.f, 0.f, 0.f};
  for (int k0 = 0; k0 < CC; k0 += 4) {
    __builtin_prefetch(arow + k0 + 64, 0, 1);  // global_prefetch_b8
    const int ak = k0 + 2 * hi;
    v2f a; a.x = arow[ak]; a.y = arow[ak + 1];
    v2f b; b.x = W[(size_t)ak * NQKV + n0 + lo];
           b.y = W[(size_t)(ak + 1) * NQKV + n0 + lo];
    c = wmma4(a, b, c);
  }
#pragma unroll
  for (int r = 0; r < 8; ++r)
    qkv[(size_t)(m0 + r + 8 * hi) * NQKV + n0 + lo] = c[r];
}

// ----------------------------- per-channel BN stats for qkv -> affine (scale,bias)
__global__ void aa_chan_stats(const float* __restrict__ qkv,
                              const float* __restrict__ gamma,
                              const float* __restrict__ beta,
                              float* __restrict__ par) {
  const int cidx = blockIdx.x;                 // 0..2047
  const int tid = threadIdx.x;
  __shared__ float s1[256], s2[256];
  float sum = 0.f, sq = 0.f;
  for (int r = tid; r < BL; r += 256) {
    float v = qkv[(size_t)r * NQKV + cidx];
    sum += v; sq += v * v;
  }
  s1[tid] = sum; s2[tid] = sq; __syncthreads();
  for (int s = 128; s > 0; s >>= 1) {
    if (tid < s) { s1[tid] += s1[tid + s]; s2[tid] += s2[tid + s]; }
    __syncthreads();
  }
  if (tid == 0) {
    float m = s1[0] * (1.f / BL);
    float v = s2[0] * (1.f / BL) - m * m;
    float rs = rsqrtf(v + EPSV);
    float a = gamma[cidx] * rs;
    par[2 * cidx] = a;
    par[2 * cidx + 1] = beta[cidx] - a * m;
  }
}

// --------------- apply BN, scatter to head layouts q/k:[B,H,L,64] v:[B,H,L,128]
__global__ void aa_qkv_scatter(const float* __restrict__ qkv,
                               const float* __restrict__ par,
                               float* __restrict__ qn, float* __restrict__ kn,
                               float* __restrict__ vn) {
  int idx = blockIdx.x * 256 + threadIdx.x;
  if (idx >= BL * NQKV) return;
  const int d = idx & (NQKV - 1);
  const int row = idx >> 11;                   // b*1024 + l
  const int b = row >> 10, l = row & 1023;
  const float v = qkv[idx] * par[2 * d] + par[2 * d + 1];
  if (d < DKX) {
    int h = d >> 6;
    qn[((size_t)(b * HH + h) * LL + l) * DKH + (d & 63)] = v;
  } else if (d < 2 * DKX) {
    int e = d - DKX; int h = e >> 6;
    kn[((size_t)(b * HH + h) * LL + l) * DKH + (e & 63)] = v;
  } else {
    int e = d - 2 * DKX; int h = e >> 7;
    vn[((size_t)(b * HH + h) * LL + l) * DVH + (e & 127)] = v;
  }
}

// -------- RPE projection GEMM: out[BHL,512] = rows[BHL,64] * emb^T[64,509pad]
// grid = (512/16, 32768/16), block = 32.
__global__ void aa_rpe_gemm(const float* __restrict__ rows,
                            const float* __restrict__ emb,
                            float* __restrict__ out) {
  const int lane = threadIdx.x & 31;
  const int hi = lane >> 4, lo = lane & 15;
  const int n0 = blockIdx.x * 16;
  const int m0 = blockIdx.y * 16;
  const int j = n0 + lo;
  const float jok = (j < NREL) ? 1.f : 0.f;
  const float* erow = emb + (size_t)((j < NREL) ? j : 0) * DKH;
  const float* arow = rows + (size_t)(m0 + lo) * DKH;
  v8f c = {0.f, 0.f, 0.f, 0.f, 0.f, 0.f, 0.f, 0.f};
  for (int k0 = 0; k0 < DKH; k0 += 4) {
    const int ak = k0 + 2 * hi;
    v2f a; a.x = arow[ak]; a.y = arow[ak + 1];
    v2f b; b.x = jok * erow[ak]; b.y = jok * erow[ak + 1];
    c = wmma4(a, b, c);
  }
#pragma unroll
  for (int r = 0; r < 8; ++r)
    out[(size_t)(m0 + r + 8 * hi) * NRELP + n0 + lo] = c[r];
}

// ------------- pass 1 over sims: accumulate per-(s,h) sum/sumsq (global atomics)
// grid = B*H*(L/16) = 2048 blocks of 256 (8 waves); wave w owns m-tiles w,w+8,...
__global__ void aa_sim_stats(const float* __restrict__ qn,
                             const float* __restrict__ kn,
                             const float* __restrict__ qr,
                             const float* __restrict__ kr,
                             float* __restrict__ acc) {
  const int blk = blockIdx.x;
  const int lt = blk & 63, h = (blk >> 6) & 7, b = blk >> 9;
  const int l0 = lt * 16;
  const int tid = threadIdx.x;
  const int wave = tid >> 5, lane = tid & 31;
  const int hi = lane >> 4, lo = lane & 15;
  __shared__ float qt[16 * DKH];
  __shared__ float red[256];
  const size_t bh = (size_t)(b * HH + h);
  stage_to_lds(qn + (bh * LL + l0) * DKH, qt, 16 * DKH, tid, 256);
  __syncthreads();
  const float* kb = kn + bh * LL * DKH;
  const float* qrb = qr + (bh * LL + l0) * NRELP;
  const float* krb = kr + bh * LL * NRELP;
  float sum[3] = {0.f, 0.f, 0.f}, sq[3] = {0.f, 0.f, 0.f};
  for (int mt = wave; mt < 64; mt += 8) {
    const int m0 = mt * 16;
    v8f c = {0.f, 0.f, 0.f, 0.f, 0.f, 0.f, 0.f, 0.f};
    for (int k0 = 0; k0 < DKH; k0 += 4) {
      const int ak = k0 + 2 * hi;
      v2f a; a.x = qt[lo * DKH + ak]; a.y = qt[lo * DKH + ak + 1];
      v2f bb; bb.x = kb[(size_t)(m0 + lo) * DKH + ak];
              bb.y = kb[(size_t)(m0 + lo) * DKH + ak + 1];
      c = wmma4(a, bb, c);
    }
#pragma unroll
    for (int r = 0; r < 8; ++r) {
      const int row = r + 8 * hi, col = lo;
      const int l = l0 + row, m = m0 + col;
      int j = m - l + 254; j = j < 0 ? 0 : (j > 508 ? 508 : j);
      const float s0 = c[r];
      const float s1 = qrb[(size_t)row * NRELP + j];
      const float s2 = krb[(size_t)m * NRELP + j];
      sum[0] += s0; sq[0] += s0 * s0;
      sum[1] += s1; sq[1] += s1 * s1;
      sum[2] += s2; sq[2] += s2 * s2;
    }
  }
  for (int s = 0; s < 3; ++s) {
    red[tid] = sum[s]; __syncthreads();
    for (int st = 128; st > 0; st >>= 1) {
      if (tid < st) red[tid] += red[tid + st];
      __syncthreads();
    }
    if (tid == 0) atomicAdd(&acc[(s * HH + h) * 2 + 0], red[0]);
    __syncthreads();
    red[tid] = sq[s]; __syncthreads();
    for (int st = 128; st > 0; st >>= 1) {
      if (tid < st) red[tid] += red[tid + st];
      __syncthreads();
    }
    if (tid == 0) atomicAdd(&acc[(s * HH + h) * 2 + 1], red[0]);
    __syncthreads();
  }
}

// ---------------- fold sims BN into per-head affine: par[h] = {a0,a1,a2,csum}
__global__ void aa_sim_finalize(const float* __restrict__ acc,
                                const float* __restrict__ gamma,
                                const float* __restrict__ beta,
                                float* __restrict__ par) {
  const int h = threadIdx.x;
  if (h >= HH) return;
  const float invN = 1.f / ((float)BB * LL * LL);
  float csum = 0.f;
  for (int s = 0; s < 3; ++s) {
    float m = acc[(s * HH + h) * 2] * invN;
    float v = acc[(s * HH + h) * 2 + 1] * invN - m * m;
    float rs = rsqrtf(v + EPSV);
    float a = gamma[s * HH + h] * rs;
    par[h * 4 + s] = a;
    csum += beta[s * HH + h] - a * m;
  }
  par[h * 4 + 3] = csum;
}

// ---------------- fused attention pass: logits -> softmax -> W*V and wbar*emb_v
// grid = 2048 blocks of 256; dynamic LDS = logits[16][1024] + qt[16][64]
//                                          + wbar[16][512] + red[272]
__global__ void aa_attention(const float* __restrict__ qn,
                             const float* __restrict__ kn,
                             const float* __restrict__ vn,
                             const float* __restrict__ qr,
                             const float* __restrict__ kr,
                             const float* __restrict__ emb_v,
                             const float* __restrict__ par,
                             float* __restrict__ r0, float* __restrict__ r1) {
  extern __shared__ float smem[];
  float* logits = smem;                        // 16*1024
  float* qt = logits + 16 * 1024;              // 16*64
  float* wbar = qt + 16 * DKH;                 // 16*512
  float* red = wbar + 16 * NRELP;              // 256 + 16
  const int blk = blockIdx.x;
  const int lt = blk & 63, h = (blk >> 6) & 7, b = blk >> 9;
  const int l0 = lt * 16;
  const int tid = threadIdx.x;
  const int wave = tid >> 5, lane = tid & 31;
  const int hi = lane >> 4, lo = lane & 15;
  const size_t bh = (size_t)(b * HH + h);
  const float a0 = par[h * 4 + 0], a1 = par[h * 4 + 1];
  const float a2 = par[h * 4 + 2], cs = par[h * 4 + 3];

  stage_to_lds(qn + (bh * LL + l0) * DKH, qt, 16 * DKH, tid, 256);
  __syncthreads();

  // Phase A: logits tile [16,1024]
  const float* kb = kn + bh * LL * DKH;
  const float* qrb = qr + (bh * LL + l0) * NRELP;
  const float* krb = kr + bh * LL * NRELP;
  for (int mt = wave; mt < 64; mt += 8) {
    const int m0 = mt * 16;
    v8f c = {0.f, 0.f, 0.f, 0.f, 0.f, 0.f, 0.f, 0.f};
    for (int k0 = 0; k0 < DKH; k0 += 4) {
      const int ak = k0 + 2 * hi;
      v2f a; a.x = qt[lo * DKH + ak]; a.y = qt[lo * DKH + ak + 1];
      v2f bb; bb.x = kb[(size_t)(m0 + lo) * DKH + ak];
              bb.y = kb[(size_t)(m0 + lo) * DKH + ak + 1];
      c = wmma4(a, bb, c);
    }
#pragma unroll
    for (int r = 0; r < 8; ++r) {
      const int row = r + 8 * hi, col = lo;
      const int l = l0 + row, m = m0 + col;
      int j = m - l + 254; j = j < 0 ? 0 : (j > 508 ? 508 : j);
      logits[row * 1024 + m] = a0 * c[r] + a1 * qrb[(size_t)row * NRELP + j] +
                               a2 * krb[(size_t)m * NRELP + j] + cs;
    }
  }
  __syncthreads();

  // Phase B: row softmax (16 threads per row)
  {
    const int row = tid >> 4, sub = tid & 15;
    float* lr = logits + row * 1024;
    float mx = -1e30f;
    for (int m = sub; m < 1024; m += 16) mx = fmaxf(mx, lr[m]);
    red[tid] = mx; __syncthreads();
    if (sub == 0) {
      float v = red[tid];
      for (int i = 1; i < 16; ++i) v = fmaxf(v, red[tid + i]);
      red[256 + row] = v;
    }
    __syncthreads();
    mx = red[256 + row];
    float sm = 0.f;
    for (int m = sub; m < 1024; m += 16) {
      float e = __expf(lr[m] - mx);
      lr[m] = e; sm += e;
    }
    __syncthreads();
    red[tid] = sm; __syncthreads();
    if (sub == 0) {
      float v = 0.f;
      for (int i = 0; i < 16; ++i) v += red[tid + i];
      red[256 + row] = 1.f / v;
    }
    __syncthreads();
    const float inv = red[256 + row];
    for (int m = sub; m < 1024; m += 16) lr[m] *= inv;
  }
  __syncthreads();

  // Phase C: scatter weights into wbar[l, rel] via LDS float atomics
  for (int i = tid; i < 16 * NRELP; i += 256) wbar[i] = 0.f;
  __syncthreads();
  for (int i = tid; i < 16 * 1024; i += 256) {
    const int row = i >> 10, m = i & 1023;
    int j = m - (l0 + row) + 254; j = j < 0 ? 0 : (j > 508 ? 508 : j);
    atomicAdd(&wbar[row * NRELP + j], logits[row * 1024 + m]);  // ds_add_f32
  }
  __syncthreads();

  // Phase D: r0 tile = weights[16,1024] * V[1024,128]; wave owns one 16-col tile
  const int d0 = wave * 16;
  const float* vb = vn + bh * LL * DVH;
  {
    v8f c = {0.f, 0.f, 0.f, 0.f, 0.f, 0.f, 0.f, 0.f};
    for (int m0 = 0; m0 < 1024; m0 += 4) {
      const int mk = m0 + 2 * hi;
      v2f a; a.x = logits[lo * 1024 + mk]; a.y = logits[lo * 1024 + mk + 1];
      v2f bb; bb.x = vb[(size_t)mk * DVH + d0 + lo];
              bb.y = vb[(size_t)(mk + 1) * DVH + d0 + lo];
      c = wmma4(a, bb, c);
    }
#pragma unroll
    for (int r = 0; r < 8; ++r)
      r0[(bh * LL + l0 + r + 8 * hi) * DVH + d0 + lo] = c[r];
  }
  // Phase E: r1 tile = wbar[16,512] * emb_v[509,128]
  {
    v8f c = {0.f, 0.f, 0.f, 0.f, 0.f, 0.f, 0.f, 0.f};
    for (int j0 = 0; j0 < NRELP; j0 += 4) {
      const int ja = j0 + 2 * hi, jb = ja + 1;
      v2f a; a.x = wbar[lo * NRELP + ja]; a.y = wbar[lo * NRELP + jb];
      v2f bb;
      bb.x = (ja < NREL) ? emb_v[(size_t)ja * DVH + d0 + lo] : 0.f;
      bb.y = (jb < NREL) ? emb_v[(size_t)jb * DVH + d0 + lo] : 0.f;
      c = wmma4(a, bb, c);
    }
#pragma unroll
    for (int r = 0; r < 8; ++r)
      r1[(bh * LL + l0 + r + 8 * hi) * DVH + d0 + lo] = c[r];
  }
}

// --------------- output BN stats per (s,h,d) over (b,l) -> affine out_par
__global__ void aa_out_stats(const float* __restrict__ r0,
                             const float* __restrict__ r1,
                             const float* __restrict__ gamma,
                             const float* __restrict__ beta,
                             float* __restrict__ par) {
  const int ch = blockIdx.x;                   // 0..2047
  const int s = ch >> 10, rem = ch & 1023, h = rem >> 7, d = rem & 127;
  const int tid = threadIdx.x;
  const float* src = (s == 0) ? r0 : r1;
  __shared__ float s1[256], s2[256];
  float sum = 0.f, sq = 0.f;
  for (int i = tid; i < BL; i += 256) {
    const int b = i >> 10, l = i & 1023;
    float v = src[((size_t)(b * HH + h) * LL + l) * DVH + d];
    sum += v; sq += v * v;
  }
  s1[tid] = sum; s2[tid] = sq; __syncthreads();
  for (int st = 128; st > 0; st >>= 1) {
    if (tid < st) { s1[tid] += s1[tid + st]; s2[tid] += s2[tid + st]; }
    __syncthreads();
  }
  if (tid == 0) {
    float m = s1[0] * (1.f / BL);
    float v = s2[0] * (1.f / BL) - m * m;
    float rs = rsqrtf(v + EPSV);
    float a = gamma[ch] * rs;
    par[2 * ch] = a;
    par[2 * ch + 1] = beta[ch] - a * m;
  }
}

// --------------- final: BN both streams, sum, write out[B,L,H*DVH]
__global__ void aa_out_final(const float* __restrict__ r0,
                             const float* __restrict__ r1,
                             const float* __restrict__ par,
                             float* __restrict__ out) {
  int idx = blockIdx.x * 256 + threadIdx.x;
  if (idx >= BB * LL * HH * DVH) return;
  const int d = idx & 127;
  int t = idx >> 7;
  const int h = t & 7; t >>= 3;
  const int l = t & 1023;
  const int b = t >> 10;
  const size_t src = ((size_t)(b * HH + h) * LL + l) * DVH + d;
  const int c0 = h * DVH + d;
  const float v0 = r0[src] * par[2 * c0] + par[2 * c0 + 1];
  const float v1 = r1[src] * par[2 * (1024 + c0)] + par[2 * (1024 + c0) + 1];
  out[idx] = v0 + v1;
}

// ---------------------------------------------------------------------------
extern "C" void kernel_launch(void* const* d_in, const int* in_sizes, int n_in,
                              void* d_out, int out_size, void* d_ws,
                              size_t ws_size, hipStream_t stream) {
  const float* x       = (const float*)d_in[0];
  const float* W_qkv   = (const float*)d_in[1];
  const float* g_qkv   = (const float*)d_in[2];
  const float* b_qkv   = (const float*)d_in[3];
  const float* g_sim   = (const float*)d_in[4];
  const float* b_sim   = (const float*)d_in[5];
  const float* g_out   = (const float*)d_in[6];
  const float* b_out   = (const float*)d_in[7];
  const float* emb_q   = (const float*)d_in[8];
  const float* emb_k   = (const float*)d_in[9];
  const float* emb_v   = (const float*)d_in[10];
  float* out = (float*)d_out;

  // workspace layout (floats); total ~58.8M floats (~235 MB)
  float* ws = (float*)d_ws;
  float* qkv  = ws;                          // 8,388,608
  float* qn   = qkv  + (size_t)BL * NQKV;    // 2,097,152
  float* kn   = qn   + (size_t)BHL * DKH;    // 2,097,152
  float* vn   = kn   + (size_t)BHL * DKH;    // 4,194,304
  float* qr   = vn   + (size_t)BHL * DVH;    // 16,777,216
  float* kr   = qr   + (size_t)BHL * NRELP;  // 16,777,216
  float* r0   = kr   + (size_t)BHL * NRELP;  // 4,194,304
  float* r1   = r0   + (size_t)BHL * DVH;    // 4,194,304
  float* chp  = r1   + (size_t)BHL * DVH;    // 4,096
  float* sacc = chp  + 2 * NQKV;             // 48
  float* spar = sacc + 48;                   // 32
  float* oacc = spar + 32;                   // 4,096 (scale/bias, 2048 ch)

  aa_zero_acc<<<16, 256, 0, stream>>>(sacc, oacc);

  aa_gemm_qkv<<<dim3(NQKV / 16, BL / 16), 32, 0, stream>>>(x, W_qkv, qkv);
  aa_chan_stats<<<NQKV, 256, 0, stream>>>(qkv, g_qkv, b_qkv, chp);
  aa_qkv_scatter<<<(BL * NQKV) / 256, 256, 0, stream>>>(qkv, chp, qn, kn, vn);

  aa_rpe_gemm<<<dim3(NRELP / 16, BHL / 16), 32, 0, stream>>>(qn, emb_q, qr);
  aa_rpe_gemm<<<dim3(NRELP / 16, BHL / 16), 32, 0, stream>>>(kn, emb_k, kr);

  aa_sim_stats<<<BB * HH * (LL / 16), 256, 0, stream>>>(qn, kn, qr, kr, sacc);
  aa_sim_finalize<<<1, 32, 0, stream>>>(sacc, g_sim, b_sim, spar);

  const size_t smem = (16 * 1024 + 16 * DKH + 16 * NRELP + 272) * sizeof(float);
  aa_attention<<<BB * HH * (LL / 16), 256, smem, stream>>>(
      qn, kn, vn, qr, kr, emb_v, spar, r0, r1);

  aa_out_stats<<<2048, 256, 0, stream>>>(r0, r1, g_out, b_out, oacc);
  aa_out_final<<<(BB * LL * HH * DVH) / 256, 256, 0, stream>>>(r0, r1, oacc,
                                                               out);
}